// GCN_44195213475900
// MI455X (gfx1250) — compile-verified
//
#include <hip/hip_runtime.h>
#include <hip/hip_bf16.h>

#define N_NODES   100000
#define N_EDGES   1250000
#define N_GRAPHS  512
#define HIDDEN    64
#define F_IN      3
#define N_CLASSES 5

typedef __attribute__((ext_vector_type(2))) float v2f;
typedef __attribute__((ext_vector_type(8))) float v8f;

// ---------------- degree: deg[dst] += 1 per edge ----------------
__global__ void gcn_degree(const int* __restrict__ dst, float* __restrict__ deg) {
    int e = blockIdx.x * blockDim.x + threadIdx.x;
    if (e < N_EDGES) atomicAdd(&deg[dst[e]], 1.0f);
}

// ---------------- dis[i] = rsqrt(deg[i] + 1)  (in place) ----------------
__global__ void gcn_deg_finalize(float* __restrict__ deg_dis) {
    int i = blockIdx.x * blockDim.x + threadIdx.x;
    if (i < N_NODES) deg_dis[i] = rsqrtf(deg_dis[i] + 1.0f);
}

// ---------------- h = x @ W1  (F_IN = 3) ----------------
__global__ void gcn_gemm1(const float* __restrict__ x, const float* __restrict__ W1,
                          float* __restrict__ h) {
    int gid = blockIdx.x * blockDim.x + threadIdx.x;
    if (gid >= N_NODES * HIDDEN) return;
    int n = gid >> 6, j = gid & 63;
    float acc = x[n * 3 + 0] * W1[0 * HIDDEN + j];
    acc = fmaf(x[n * 3 + 1], W1[1 * HIDDEN + j], acc);
    acc = fmaf(x[n * 3 + 2], W1[2 * HIDDEN + j], acc);
    h[gid] = acc;
}

// ---------------- edge scatter: agg[dst] += h[src] * dis[src]*dis[dst] ----------------
__global__ void gcn_scatter(const int* __restrict__ src, const int* __restrict__ dst,
                            const float* __restrict__ dis, const float* __restrict__ h,
                            float* __restrict__ agg) {
    long long gid = (long long)blockIdx.x * blockDim.x + threadIdx.x;
    if (gid >= (long long)N_EDGES * HIDDEN) return;
    int e = (int)(gid >> 6);
    int j = (int)(gid & 63);
    int s = src[e], d = dst[e];
    float norm = dis[s] * dis[d];
    atomicAdd(&agg[(long long)d * HIDDEN + j], h[(long long)s * HIDDEN + j] * norm);
}

// ---------------- h = act(agg + h*dis^2 + b)  (in place on h) ----------------
__global__ void gcn_fuse(const float* __restrict__ agg, const float* __restrict__ dis,
                         const float* __restrict__ b, float* __restrict__ h, int do_relu) {
    int gid = blockIdx.x * blockDim.x + threadIdx.x;
    if (gid >= N_NODES * HIDDEN) return;
    int n = gid >> 6, j = gid & 63;
    float s = dis[n];
    float v = agg[gid] + h[gid] * (s * s) + b[j];
    h[gid] = do_relu ? fmaxf(v, 0.0f) : v;
}

// ---------------- O[100000,64] = H[100000,64] @ W[64,64]  via V_WMMA_F32_16X16X4_F32 ----
// 320 threads = 10 waves per block; each wave owns one 16-row tile (6250 tiles total,
// 625 blocks * 10 waves exactly, so EXEC stays all-1s for every WMMA).
__global__ void __launch_bounds__(320)
gcn_gemm2_wmma(const float* __restrict__ H, const float* __restrict__ W,
               float* __restrict__ O) {
    __shared__ float Ws[HIDDEN * HIDDEN];          // 16 KB of W2 in LDS
    for (int i = threadIdx.x; i < HIDDEN * HIDDEN; i += 320) Ws[i] = W[i];
    __syncthreads();

    const int wave = threadIdx.x >> 5;             // 0..9
    const int lane = threadIdx.x & 31;
    const int rowTile = blockIdx.x * 10 + wave;    // 0..6249
    const int row0 = rowTile * 16;
    const int m  = lane & 15;                      // row (A) / col (B,C,D) within tile
    const int kh = lane >> 4;                      // 0: K=0,1   1: K=2,3   (A/B frag halves)

    v8f acc0 = {}, acc1 = {}, acc2 = {}, acc3 = {};
    const float* hrow = H + (long long)(row0 + m) * HIDDEN;

    for (int kk = 0; kk < HIDDEN / 4; ++kk) {
        const int kbase = kk * 4 + kh * 2;
        v2f a;
        a.x = hrow[kbase + 0];
        a.y = hrow[kbase + 1];

        v2f b0, b1, b2, b3;
        b0.x = Ws[(kbase + 0) * HIDDEN +  0 + m];  b0.y = Ws[(kbase + 1) * HIDDEN +  0 + m];
        b1.x = Ws[(kbase + 0) * HIDDEN + 16 + m];  b1.y = Ws[(kbase + 1) * HIDDEN + 16 + m];
        b2.x = Ws[(kbase + 0) * HIDDEN + 32 + m];  b2.y = Ws[(kbase + 1) * HIDDEN + 32 + m];
        b3.x = Ws[(kbase + 0) * HIDDEN + 48 + m];  b3.y = Ws[(kbase + 1) * HIDDEN + 48 + m];

        acc0 = __builtin_amdgcn_wmma_f32_16x16x4_f32(false, a, false, b0, (short)0, acc0, false, false);
        acc1 = __builtin_amdgcn_wmma_f32_16x16x4_f32(false, a, false, b1, (short)0, acc1, false, false);
        acc2 = __builtin_amdgcn_wmma_f32_16x16x4_f32(false, a, false, b2, (short)0, acc2, false, false);
        acc3 = __builtin_amdgcn_wmma_f32_16x16x4_f32(false, a, false, b3, (short)0, acc3, false, false);
    }

    // C/D layout: VGPR i -> M = i (lanes 0-15) or i+8 (lanes 16-31); N = lane&15
    #pragma unroll
    for (int i = 0; i < 8; ++i) {
        long long r = (long long)(row0 + i + kh * 8) * HIDDEN;
        O[r +  0 + m] = acc0[i];
        O[r + 16 + m] = acc1[i];
        O[r + 32 + m] = acc2[i];
        O[r + 48 + m] = acc3[i];
    }
}

// ---------------- mean-pool accumulation ----------------
__global__ void gcn_pool(const int* __restrict__ batch, const float* __restrict__ h,
                         float* __restrict__ sums, float* __restrict__ cnts) {
    int gid = blockIdx.x * blockDim.x + threadIdx.x;
    if (gid >= N_NODES * HIDDEN) return;
    int n = gid >> 6, j = gid & 63;
    int g = batch[n];
    atomicAdd(&sums[g * HIDDEN + j], h[gid]);
    if (j == 0) atomicAdd(&cnts[g], 1.0f);
}

// ---------------- out = (sums/cnt) @ Wl + bl ----------------
__global__ void gcn_final(const float* __restrict__ sums, const float* __restrict__ cnts,
                          const float* __restrict__ Wl, const float* __restrict__ bl,
                          float* __restrict__ out) {
    int gid = blockIdx.x * blockDim.x + threadIdx.x;
    if (gid >= N_GRAPHS * N_CLASSES) return;
    int g = gid / N_CLASSES, c = gid % N_CLASSES;
    float inv = 1.0f / fmaxf(cnts[g], 1.0f);
    float acc = bl[c];
    #pragma unroll 8
    for (int j = 0; j < HIDDEN; ++j)
        acc = fmaf(sums[g * HIDDEN + j] * inv, Wl[j * N_CLASSES + c], acc);
    out[gid] = acc;
}

extern "C" void kernel_launch(void* const* d_in, const int* in_sizes, int n_in,
                              void* d_out, int out_size, void* d_ws, size_t ws_size,
                              hipStream_t stream) {
    const float* x    = (const float*)d_in[0];
    const int*   ei   = (const int*)d_in[1];       // [2, N_EDGES]: row0=src, row1=dst
    const int*   batch= (const int*)d_in[2];
    const float* W1   = (const float*)d_in[3];
    const float* b1   = (const float*)d_in[4];
    const float* W2   = (const float*)d_in[5];
    const float* b2   = (const float*)d_in[6];
    const float* Wl   = (const float*)d_in[7];
    const float* bl   = (const float*)d_in[8];
    float*       out  = (float*)d_out;

    const int* src = ei;
    const int* dst = ei + N_EDGES;

    // workspace carve-up (floats)
    float* ws   = (float*)d_ws;
    float* dis  = ws;                                  // N_NODES (deg, then rsqrt in place)
    float* h    = dis + N_NODES;                       // N_NODES*64 (h1 in place)
    float* agg  = h   + (size_t)N_NODES * HIDDEN;      // N_NODES*64
    float* h2   = agg + (size_t)N_NODES * HIDDEN;      // N_NODES*64
    float* sums = h2  + (size_t)N_NODES * HIDDEN;      // 512*64
    float* cnts = sums + N_GRAPHS * HIDDEN;            // 512

    const int NF  = N_NODES * HIDDEN;                  // 6.4M elementwise work items
    const long long EF = (long long)N_EDGES * HIDDEN;  // 80M scatter work items
    const int TB = 256;

    // zero accumulators (stream-ordered, capture-safe)
    hipMemsetAsync(dis,  0, (size_t)N_NODES * sizeof(float), stream);
    hipMemsetAsync(agg,  0, (size_t)NF * sizeof(float), stream);
    hipMemsetAsync(sums, 0, (size_t)(N_GRAPHS * HIDDEN + N_GRAPHS) * sizeof(float), stream);

    gcn_degree<<<(N_EDGES + TB - 1) / TB, TB, 0, stream>>>(dst, dis);
    gcn_deg_finalize<<<(N_NODES + TB - 1) / TB, TB, 0, stream>>>(dis);

    // layer 1
    gcn_gemm1<<<(NF + TB - 1) / TB, TB, 0, stream>>>(x, W1, h);
    gcn_scatter<<<(unsigned)((EF + TB - 1) / TB), TB, 0, stream>>>(src, dst, dis, h, agg);
    gcn_fuse<<<(NF + TB - 1) / TB, TB, 0, stream>>>(agg, dis, b1, h, /*relu=*/1);

    // layer 2
    hipMemsetAsync(agg, 0, (size_t)NF * sizeof(float), stream);
    gcn_gemm2_wmma<<<625, 320, 0, stream>>>(h, W2, h2);
    gcn_scatter<<<(unsigned)((EF + TB - 1) / TB), TB, 0, stream>>>(src, dst, dis, h2, agg);
    gcn_fuse<<<(NF + TB - 1) / TB, TB, 0, stream>>>(agg, dis, b2, h2, /*relu=*/0);

    // pool + classifier
    gcn_pool<<<(NF + TB - 1) / TB, TB, 0, stream>>>(batch, h2, sums, cnts);
    gcn_final<<<(N_GRAPHS * N_CLASSES + TB - 1) / TB, TB, 0, stream>>>(sums, cnts, Wl, bl, out);
}